// Log_Cosin_Loss_67577015435611
// MI455X (gfx1250) — compile-verified
//
#include <hip/hip_runtime.h>
#include <math.h>

// D = A*B + C on f32 16x16x4 WMMA. A frag: lane l holds row (l%16),
// K = 2*(l>>4) + {0,1} in its 2 VGPRs. B frag is the transpose-symmetric
// layout, so a fragment loaded with identical per-lane indexing from a
// row-major matrix acts as A for that matrix and as B for its transpose.
typedef __attribute__((ext_vector_type(2))) float v2f;
typedef __attribute__((ext_vector_type(8))) float v8f;

#define DIMS 1024
#define ROWS_PER_BLOCK 16
#define WAVES_PER_BLOCK 8
#define COLS_PER_WAVE (DIMS / WAVES_PER_BLOCK) /* 128 */
#define CHUNKS_PER_WAVE (COLS_PER_WAVE / 4)    /* 32  */

__global__ __launch_bounds__(256) void cosine_diag_partial_kernel(
    const float* __restrict__ x, const float* __restrict__ y,
    float* __restrict__ partial) {
  const int lane = threadIdx.x & 31;
  const int wave = threadIdx.x >> 5;
  const int half = lane >> 4;   // which K half of the fragment
  const int mrow = lane & 15;   // row within the 16-row tile

  const size_t row = (size_t)blockIdx.x * ROWS_PER_BLOCK + mrow;
  const float* xr = x + row * DIMS;
  const float* yr = y + row * DIMS;
  const int colbase = wave * COLS_PER_WAVE + 2 * half;

  v8f cxy = {};  // D[m][n] += sum_k x[m][k] * y[n][k]  (diag = xy dots)
  v8f cxx = {};  // gram of x (diag = ||x_i||^2)
  v8f cyy = {};  // gram of y (diag = ||y_i||^2)

#pragma unroll 4
  for (int j = 0; j < CHUNKS_PER_WAVE; ++j) {
    const int c = colbase + 4 * j;          // even -> 8B aligned
    const v2f ax = *(const v2f*)(xr + c);   // global_load_b64
    const v2f ay = *(const v2f*)(yr + c);
    cxy = __builtin_amdgcn_wmma_f32_16x16x4_f32(false, ax, false, ay,
                                                (short)0, cxy, false, false);
    cxx = __builtin_amdgcn_wmma_f32_16x16x4_f32(false, ax, false, ax,
                                                (short)0, cxx, false, false);
    cyy = __builtin_amdgcn_wmma_f32_16x16x4_f32(false, ay, false, ay,
                                                (short)0, cyy, false, false);
  }

  // C/D layout: lanes 0-15: (M=v, N=lane); lanes 16-31: (M=v+8, N=lane-16).
  // Diagonal row r<8 -> c[r] on lane r; row r>=8 -> c[r-8] on lane r+16.
  float dxy = 0.f, dxx = 0.f, dyy = 0.f;
#pragma unroll
  for (int v = 0; v < 8; ++v) {
    if (lane == v || lane == v + 24) {
      dxy = cxy[v]; dxx = cxx[v]; dyy = cyy[v];
    }
  }

  __shared__ float sxy[WAVES_PER_BLOCK][ROWS_PER_BLOCK];
  __shared__ float sxx[WAVES_PER_BLOCK][ROWS_PER_BLOCK];
  __shared__ float syy[WAVES_PER_BLOCK][ROWS_PER_BLOCK];
  __shared__ float sloss[ROWS_PER_BLOCK];

  if (lane < 8 || lane >= 24) {
    const int r = (lane < 8) ? lane : (lane - 16);
    sxy[wave][r] = dxy; sxx[wave][r] = dxx; syy[wave][r] = dyy;
  }
  __syncthreads();

  if (threadIdx.x < ROWS_PER_BLOCK) {
    const int r = threadIdx.x;
    float xy = 0.f, xx = 0.f, yy = 0.f;
#pragma unroll
    for (int w = 0; w < WAVES_PER_BLOCK; ++w) {
      xy += sxy[w][r]; xx += sxx[w][r]; yy += syy[w][r];
    }
    const float nx = fmaxf(sqrtf(xx), 1e-12f);
    const float ny = fmaxf(sqrtf(yy), 1e-12f);
    const float cosv = xy / (nx * ny);
    sloss[r] = -logf((cosv + 1.0f) * 0.5f);
  }
  __syncthreads();

  if (threadIdx.x == 0) {
    float s = 0.f;
#pragma unroll
    for (int r = 0; r < ROWS_PER_BLOCK; ++r) s += sloss[r];
    partial[blockIdx.x] = s;  // deterministic per-block partial
  }
}

__global__ __launch_bounds__(256) void reduce_mean_kernel(
    const float* __restrict__ partial, int nblocks, int nrows,
    float* __restrict__ out) {
  __shared__ float s[256];
  float v = 0.f;
  for (int i = threadIdx.x; i < nblocks; i += 256) v += partial[i];
  s[threadIdx.x] = v;
  __syncthreads();
  for (int off = 128; off > 0; off >>= 1) {
    if ((int)threadIdx.x < off) s[threadIdx.x] += s[threadIdx.x + off];
    __syncthreads();
  }
  if (threadIdx.x == 0) out[0] = s[0] / (float)nrows;
}

extern "C" void kernel_launch(void* const* d_in, const int* in_sizes, int n_in,
                              void* d_out, int out_size, void* d_ws, size_t ws_size,
                              hipStream_t stream) {
  (void)n_in; (void)out_size; (void)ws_size;
  const float* x = (const float*)d_in[0];
  const float* y = (const float*)d_in[1];
  float* out = (float*)d_out;
  float* partial = (float*)d_ws;  // nblocks floats of scratch

  const int N = in_sizes[0] / DIMS;        // 8192
  const int nblocks = N / ROWS_PER_BLOCK;  // 512

  cosine_diag_partial_kernel<<<dim3(nblocks), dim3(256), 0, stream>>>(x, y, partial);
  reduce_mean_kernel<<<dim3(1), dim3(256), 0, stream>>>(partial, nblocks, N, out);
}